// MLPPredictor_75213467287860
// MI455X (gfx1250) — compile-verified
//
#include <hip/hip_runtime.h>
#include <hip/hip_bf16.h>

typedef float v2f __attribute__((ext_vector_type(2)));
typedef float v8f __attribute__((ext_vector_type(8)));

#define D_FEAT 128   // feature dim
#define C_OUT  2     // classes

// ---------------------------------------------------------------------------
// Kernel 1: per-node projection p[n, 0..3] = [h·Wu0, h·Wu1, h·Wv0, h·Wv1]
// via V_WMMA_F32_16X16X4_F32. One wave = one 16-node tile, K=128 in 32 steps.
//
// A (16x4 f32, 2 VGPR): lanes 0-15 hold (M=lane, K=k..k+1); lanes 16-31 hold
//                       (M=lane-16, K=k+2..k+3).
// B (4x16 f32, 2 VGPR): lanes 0-15 hold (K=k..k+1, N=lane); lanes 16-31 hold
//                       (K=k+2..k+3, N=lane-16). Columns N>=4 are zero.
// D (16x16 f32, 8 VGPR): VGPR r: lanes 0-15 -> (M=r, N=lane),
//                                 lanes 16-31 -> (M=r+8, N=lane-16).
// ---------------------------------------------------------------------------
__global__ __launch_bounds__(128) void gnn_node_proj_wmma(
    const float* __restrict__ h,
    const float* __restrict__ W,     // (2, 256): [Wu0 Wv0 ; Wu1 Wv1]
    float* __restrict__ p,           // (n_tiles*16, 4)
    int n_nodes, int n_tiles)
{
  const int lane   = threadIdx.x & 31;
  const int tile   = blockIdx.x * 4 + (threadIdx.x >> 5);
  if (tile >= n_tiles) return;                 // wave-uniform: EXEC stays all-1s

  const int laneLo = lane & 15;
  const int kHalf  = (lane >> 4) * 2;          // lanes 16-31 own K+2..K+3

  // A-matrix source row (clamped for a possible partial last tile)
  int row = tile * 16 + laneLo;
  if (row >= n_nodes) row = n_nodes - 1;
  const float* aPtr = h + (size_t)row * D_FEAT + kHalf;

  // B-matrix column n: 0,1 -> W_u rows 0,1 ; 2,3 -> W_v rows 0,1 ; >=4 -> zero
  const int  n      = laneLo;
  const bool nValid = (n < 2 * C_OUT);
  const int  wOff   = nValid ? ((n & 1) * (2 * D_FEAT) + (n >> 1) * D_FEAT) : 0;
  const float* bPtr = W + wOff + kHalf;

  v8f acc = {};
#pragma unroll
  for (int k = 0; k < D_FEAT; k += 4) {
    v2f a  = *(const v2f*)(aPtr + k);          // global_load_b64
    v2f bv = *(const v2f*)(bPtr + k);          // L1-resident (W = 2 KB)
    if (!nValid) { bv.x = 0.0f; bv.y = 0.0f; } // zero-pad columns 4..15
    acc = __builtin_amdgcn_wmma_f32_16x16x4_f32(
        /*neg_a=*/false, a, /*neg_b=*/false, bv,
        /*c_mod=*/(short)0, acc, /*reuse_a=*/false, /*reuse_b=*/false);
  }

  // Scatter D: only columns n<4 are real outputs.
  if (nValid) {
    float* dptr = p + (size_t)(tile * 16 + (lane >> 4) * 8) * 4 + n;
#pragma unroll
    for (int r = 0; r < 8; ++r)
      dptr[(size_t)r * 4] = acc[r];
  }
}

// ---------------------------------------------------------------------------
// Kernel 2: out[e,c] = p[src[e],c] + p[dst[e],2+c] + b[c]
// 8-byte gathers from an 800 KB L2-resident table; streaming index reads.
// ---------------------------------------------------------------------------
__global__ __launch_bounds__(256) void gnn_edge_score(
    const float* __restrict__ p,
    const int* __restrict__ src,
    const int* __restrict__ dst,
    const float* __restrict__ b,
    float* __restrict__ out,
    int n_edges)
{
  const int e = blockIdx.x * blockDim.x + threadIdx.x;
  if (e >= n_edges) return;
  const float b0 = b[0];
  const float b1 = b[1];
  const int s = src[e];
  const int d = dst[e];
  const v2f pu = *(const v2f*)(p + (size_t)s * 4);       // c = 0,1 (W_u part)
  const v2f pv = *(const v2f*)(p + (size_t)d * 4 + 2);   // c = 0,1 (W_v part)
  v2f o;
  o.x = pu.x + pv.x + b0;
  o.y = pu.y + pv.y + b1;
  *(v2f*)(out + (size_t)e * 2) = o;                      // global_store_b64
}

// ---------------------------------------------------------------------------
// Launch: inputs in reference order: h, W, b, src, dst
// ---------------------------------------------------------------------------
extern "C" void kernel_launch(void* const* d_in, const int* in_sizes, int n_in,
                              void* d_out, int out_size, void* d_ws, size_t ws_size,
                              hipStream_t stream) {
  const float* h   = (const float*)d_in[0];
  const float* W   = (const float*)d_in[1];
  const float* b   = (const float*)d_in[2];
  const int*   src = (const int*)d_in[3];
  const int*   dst = (const int*)d_in[4];
  float*       out = (float*)d_out;
  float*       p   = (float*)d_ws;   // n_tiles*16*4 floats (800 KB for 50 K nodes)

  const int n_nodes = in_sizes[0] / D_FEAT;
  const int n_edges = in_sizes[3];
  const int n_tiles = (n_nodes + 15) / 16;

  // 4 waves/block, 1 tile (16 nodes) per wave
  const int grd1 = (n_tiles + 3) / 4;
  gnn_node_proj_wmma<<<dim3(grd1), dim3(128), 0, stream>>>(h, W, p, n_nodes, n_tiles);

  const int grd2 = (n_edges + 255) / 256;
  gnn_edge_score<<<dim3(grd2), dim3(256), 0, stream>>>(p, src, dst, b, out, n_edges);
}